// MobiusGRU_53558242181268
// MI455X (gfx1250) — compile-verified
//
#include <hip/hip_runtime.h>
#include <hip/hip_bf16.h>
#include <math.h>

#define SEQL  1024
#define BATCH 64
#define DIMS  256
#define HID   256
#define NL    2
#define G3    (3*HID)           // 768
#define NROWS (SEQL*BATCH)      // 65536
#define EPSF  1e-7f
#define EPS2F (1e-7f*1e-7f)

typedef float v2f __attribute__((ext_vector_type(2)));
typedef float v8f __attribute__((ext_vector_type(8)));

// Persistent scratch in device BSS (independent of ws_size, graph-capture safe,
// rewritten deterministically every call).
__device__ float  g_ux[(size_t)NROWS * G3];  // hoisted x@U^T (row-scaled), 192 MB
__device__ float  g_mxnu[NROWS * 3];         // per-row per-chunk ||Ux|| (clamped)
__device__ float  g_s[NROWS];                // expmap0 row scale (layer 0)
__device__ float  g_xn[NROWS];               // ||x|| per row (on-ball input norm)
__device__ float2 g_wpk[3*128*HID];          // w_hh repacked K-pair-major (768 KB)

// ---------------- wave32 helpers ----------------
__device__ __forceinline__ float wsum(float v){
  v += __shfl_xor(v, 1, 32);
  v += __shfl_xor(v, 2, 32);
  v += __shfl_xor(v, 4, 32);
  v += __shfl_xor(v, 8, 32);
  v += __shfl_xor(v, 16, 32);
  return v;
}
__device__ __forceinline__ float rdot8(const float* a, const float* b){
  float p = 0.f;
#pragma unroll
  for (int j = 0; j < 8; ++j) p += a[j]*b[j];
  return wsum(p);
}
__device__ __forceinline__ float artanh_(float x){
  x = fminf(fmaxf(x, -1.f + 1e-7f), 1.f - 1e-7f);
  return 0.5f*(log1pf(x) - log1pf(-x));
}
// o = mobius_add(x, y) given raw squared norms x2, y2 (C = 1)
__device__ __forceinline__ void mobius_add8(const float* x, const float* y,
                                            float x2, float y2, float* o){
  float xy  = rdot8(x, y);
  float cx  = 1.f + 2.f*xy + y2;
  float cy  = 1.f - x2;
  float den = fmaxf(1.f + 2.f*xy + x2*y2, EPSF);
  float rd  = 1.f/den;
#pragma unroll
  for (int j = 0; j < 8; ++j) o[j] = (cx*x[j] + cy*y[j]) * rd;
}

// g = mobius_add(mobius_add(mobius(Wh), mobius(Ux)), bias)  (one_rnn_transform)
__device__ __forceinline__ void rnn_transform_row(const float* wh, float whd,
                                                  float bn, float abn,
                                                  const float* u, float un,
                                                  float xnr, float axn,
                                                  const float* bc, float b2c,
                                                  float* g){
  float whn = sqrtf(fmaxf(whd, EPS2F));
  float sW  = tanhf(whn/bn * abn)/whn;        // mobius_matvec tail on Wh
  float sU  = tanhf(un/xnr * axn)/un;         // mobius_matvec tail on Ux
  float whm[8], um[8];
#pragma unroll
  for (int j = 0; j < 8; ++j){ whm[j] = sW*wh[j]; um[j] = sU*u[j]; }
  float x2 = sW*sW*whd;
  float y2 = sU*sU*un*un;
  float a[8];
  mobius_add8(whm, um, x2, y2, a);
  float a2 = rdot8(a, a);
  mobius_add8(a, bc, a2, b2c, g);
}

// ---------------- weight repack: w_hh -> K-pair-major float2 panels ----------
// g_wpk[(c*128 + k/2)*HID + n] = { W[c*256+k][n], W[c*256+k+1][n] }
__global__ __launch_bounds__(256) void pack_whh(const float* __restrict__ whh){
  int i = blockIdx.x*256 + threadIdx.x;        // < 3*128*256 = 98304
  int n  = i & 255;
  int kp = (i >> 8) & 127;
  int c  = i >> 15;
  size_t krow = (size_t)(c*256 + 2*kp);
  g_wpk[i] = make_float2(whh[krow*HID + n], whh[(krow+1)*HID + n]);
}

// ---------------- hoisted input projection ----------------
// g_ux[row, 0:768] = scale(row) * X[row,:] @ U^T  via V_WMMA_F32_16X16X4_F32
__global__ __launch_bounds__(512) void ux_gemm(const float* __restrict__ X,
                                               const float* __restrict__ U,
                                               int use_scale){
  const int lane = threadIdx.x & 31;
  const int job  = blockIdx.x*16 + (threadIdx.x >> 5);    // 4096*48 jobs
  const int mt = job / 48, nt = job % 48;
  const int m0 = mt*16, n0 = nt*16;
  const int m  = lane & 15;
  const int kh = lane >> 4;
  const float2* xr = (const float2*)(X + (size_t)(m0 + m)*DIMS);
  const float2* ur = (const float2*)(U + (size_t)(n0 + m)*DIMS);
  v8f acc = {0.f,0.f,0.f,0.f,0.f,0.f,0.f,0.f};
#pragma unroll 4
  for (int k0 = 0; k0 < DIMS; k0 += 4){
    float2 af = xr[(k0 >> 1) + kh];            // K = k0 + 2*kh (+1)
    float2 bf = ur[(k0 >> 1) + kh];
    v2f a, b;
    a.x = af.x; a.y = af.y;
    b.x = bf.x; b.y = bf.y;
    acc = __builtin_amdgcn_wmma_f32_16x16x4_f32(false, a, false, b, (short)0,
                                                acc, false, false);
  }
  if (use_scale){                              // uniform branch, hoisted
#pragma unroll
    for (int i = 0; i < 8; ++i){
      int R = m0 + kh*8 + i;
      g_ux[(size_t)R*G3 + n0 + m] = g_s[R] * acc[i];
    }
  } else {
#pragma unroll
    for (int i = 0; i < 8; ++i){
      int R = m0 + kh*8 + i;
      g_ux[(size_t)R*G3 + n0 + m] = acc[i];
    }
  }
}

// ---------------- norm precompute ----------------
__global__ __launch_bounds__(256) void row_expmap_norms(const float* __restrict__ X){
  const int lane = threadIdx.x & 31;
  const size_t R = (size_t)blockIdx.x*8 + (threadIdx.x >> 5);
  float p = 0.f;
#pragma unroll
  for (int j = 0; j < 8; ++j){ float v = X[R*DIMS + lane + 32*j]; p += v*v; }
  p = wsum(p);
  float un = sqrtf(fmaxf(p, EPS2F));
  float th = tanhf(un);
  if (lane == 0){ g_s[R] = th/un; g_xn[R] = th; }   // ||expmap0(u)|| = tanh||u||
}

__global__ __launch_bounds__(256) void ux_norms(const float* __restrict__ X){
  const int lane = threadIdx.x & 31;
  const size_t R = (size_t)blockIdx.x*8 + (threadIdx.x >> 5);
  const float* yr = g_ux + R*G3;
#pragma unroll
  for (int c = 0; c < 3; ++c){
    float p = 0.f;
#pragma unroll
    for (int j = 0; j < 8; ++j){ float v = yr[c*HID + lane + 32*j]; p += v*v; }
    p = wsum(p);
    if (lane == 0) g_mxnu[R*3 + c] = sqrtf(fmaxf(p, EPS2F));
  }
  if (X){  // layer 1: input-row norms come from layer-0 outputs
    float p = 0.f;
#pragma unroll
    for (int j = 0; j < 8; ++j){ float v = X[R*HID + lane + 32*j]; p += v*v; }
    p = wsum(p);
    if (lane == 0) g_xn[R] = sqrtf(fmaxf(p, EPS2F));
  }
}

// ---------------- the sequential Mobius-GRU scan ----------------
// 4 independent workgroups (batch slices of 16); 16 waves: wave==batch row for
// elementwise phases, wave==N-tile for WMMA phases. h/rh in XOR-swizzled LDS.
// Per step: [dual-GEMM z+r] -> [gates + rh] -> [GEMM candidate] -> [update].
__global__ __launch_bounds__(512) void mobius_gru_scan(
    const float* __restrict__ h0,     // [BATCH, HID]
    const float* __restrict__ bias,   // [3, HID]
    float* __restrict__ outp,         // [SEQ*BATCH, HID]
    float* __restrict__ hlast)        // [BATCH, HID]
{
  __shared__ float hbuf [16*HID];     // swizzled (A-fragment source)
  __shared__ float rhbuf[16*HID];     // swizzled (A-fragment source)
  __shared__ float tmpz [16*HID];     // GEMM staging (z / candidate)
  __shared__ float tmpr [16*HID];     // GEMM staging (r)
  const int lane = threadIdx.x & 31;
  const int row  = threadIdx.x >> 5;          // 0..15
  const int bg   = blockIdx.x*16 + row;       // global batch row
  const int sw   = row << 2;
  // WMMA-phase constants
  const int m  = lane & 15;
  const int kh = lane >> 4;
  const int n0 = row * 16;                    // wave's N tile
  const int swm = m << 2;
  const float2* wr = g_wpk;                   // chunk r
  const float2* wc = g_wpk + (size_t)1*128*HID; // chunk h (candidate)
  const float2* wz = g_wpk + (size_t)2*128*HID; // chunk z

  float br[8], bh[8], bz[8];
#pragma unroll
  for (int j = 0; j < 8; ++j){
    int c = lane + 32*j;
    br[j] = bias[0*HID + c]; bh[j] = bias[1*HID + c]; bz[j] = bias[2*HID + c];
  }
  const float b2r = rdot8(br,br), b2h = rdot8(bh,bh), b2z = rdot8(bz,bz);

  float h[8];
#pragma unroll
  for (int j = 0; j < 8; ++j){
    int c = lane + 32*j;
    h[j] = h0[(size_t)bg*HID + c];
    hbuf[row*HID + (c ^ sw)] = h[j];
  }
  __syncthreads();

  for (int t = 0; t < SEQL; ++t){
    const size_t rix = (size_t)t*BATCH + bg;
    const float* uxr = g_ux   + rix*G3;
    const float* mu3 = g_mxnu + rix*3;
    const float xnr = g_xn[rix];
    const float axn = artanh_(xnr);
    const float h2  = rdot8(h, h);
    const float hn  = sqrtf(fmaxf(h2, EPS2F));
    const float ahn = artanh_(hn);

    // ---- dual GEMM: Wh_z = h@W_hz and Wh_r = h@W_hr (shared A fragments)
    {
      v8f az = {0.f,0.f,0.f,0.f,0.f,0.f,0.f,0.f};
      v8f ar = {0.f,0.f,0.f,0.f,0.f,0.f,0.f,0.f};
#pragma unroll 4
      for (int k0 = 0; k0 < HID; k0 += 4){
        int k = k0 + 2*kh;
        v2f a = *(const v2f*)(hbuf + m*HID + (k ^ swm));  // ds_load_b64
        float2 bzf = wz[(size_t)((k0 >> 1) + kh)*HID + n0 + m];
        float2 brf = wr[(size_t)((k0 >> 1) + kh)*HID + n0 + m];
        v2f b1; b1.x = bzf.x; b1.y = bzf.y;
        v2f b2; b2.x = brf.x; b2.y = brf.y;
        az = __builtin_amdgcn_wmma_f32_16x16x4_f32(false, a, false, b1, (short)0,
                                                   az, false, false);
        ar = __builtin_amdgcn_wmma_f32_16x16x4_f32(false, a, false, b2, (short)0,
                                                   ar, false, false);
      }
#pragma unroll
      for (int i = 0; i < 8; ++i){             // C: M=(lane/16)*8+i, N=lane%16
        tmpz[(kh*8 + i)*HID + n0 + m] = az[i];
        tmpr[(kh*8 + i)*HID + n0 + m] = ar[i];
      }
    }
    __syncthreads();

    // ---- gates z, r and rh = mobius_pointwise_mul(r, h)   (wave == row)
    float zv[8], rh[8], rhn;
    {
      float whz[8], whr[8], uz[8], ur[8];
#pragma unroll
      for (int j = 0; j < 8; ++j){
        int c = lane + 32*j;
        whz[j] = tmpz[row*HID + c];
        whr[j] = tmpr[row*HID + c];
        uz[j]  = uxr[2*HID + c];
        ur[j]  = uxr[0*HID + c];
      }
      // z gate
      float g[8];
      rnn_transform_row(whz, rdot8(whz,whz), hn, ahn, uz, mu3[2], xnr, axn, bz, b2z, g);
      float gn = sqrtf(fmaxf(rdot8(g,g), EPS2F));
      float lm = artanh_(gn)/gn;                     // logmap0
#pragma unroll
      for (int j = 0; j < 8; ++j) zv[j] = 1.f/(1.f + expf(-lm*g[j]));
      // r gate
      rnn_transform_row(whr, rdot8(whr,whr), hn, ahn, ur, mu3[0], xnr, axn, br, b2r, g);
      gn = sqrtf(fmaxf(rdot8(g,g), EPS2F));
      lm = artanh_(gn)/gn;
      float m_[8];
#pragma unroll
      for (int j = 0; j < 8; ++j){
        float rv = 1.f/(1.f + expf(-lm*g[j]));
        m_[j] = rv * h[j];
      }
      float md = rdot8(m_, m_);
      float mn = sqrtf(fmaxf(md, EPS2F));
      float sc = tanhf(mn/hn*ahn)/mn;
#pragma unroll
      for (int j = 0; j < 8; ++j){
        rh[j] = sc*m_[j];
        rhbuf[row*HID + ((lane + 32*j) ^ sw)] = rh[j];
      }
      rhn = sqrtf(fmaxf(sc*sc*md, EPS2F));
      // prefetch next timestep's Ux row while the candidate GEMM runs
      if (t + 1 < SEQL)
        __builtin_prefetch(uxr + (size_t)BATCH*G3 + lane*32, 0, 3);
    }
    __syncthreads();

    // ---- candidate GEMM: Wh_h = rh @ W_hh_ (chunk 1)
    {
      v8f ac = {0.f,0.f,0.f,0.f,0.f,0.f,0.f,0.f};
#pragma unroll 4
      for (int k0 = 0; k0 < HID; k0 += 4){
        int k = k0 + 2*kh;
        v2f a = *(const v2f*)(rhbuf + m*HID + (k ^ swm));
        float2 bf = wc[(size_t)((k0 >> 1) + kh)*HID + n0 + m];
        v2f b; b.x = bf.x; b.y = bf.y;
        ac = __builtin_amdgcn_wmma_f32_16x16x4_f32(false, a, false, b, (short)0,
                                                   ac, false, false);
      }
#pragma unroll
      for (int i = 0; i < 8; ++i)
        tmpz[(kh*8 + i)*HID + n0 + m] = ac[i];
    }
    __syncthreads();

    // ---- h_tilde, delta, gated update   (wave == row; z kept in registers)
    {
      float arhn = artanh_(rhn);
      float wh[8], u[8];
#pragma unroll
      for (int j = 0; j < 8; ++j){
        wh[j] = tmpz[row*HID + lane + 32*j];
        u[j]  = uxr[1*HID + lane + 32*j];
      }
      float ht_[8];
      rnn_transform_row(wh, rdot8(wh,wh), rhn, arhn, u, mu3[1], xnr, axn, bh, b2h, ht_);
      float ht2 = rdot8(ht_, ht_);
      float negh[8];
#pragma unroll
      for (int j = 0; j < 8; ++j) negh[j] = -h[j];
      float delta[8];
      mobius_add8(negh, ht_, h2, ht2, delta);        // delta = (-h) (+) h_tilde
      float d2 = rdot8(delta, delta);
      float dn = sqrtf(fmaxf(d2, EPS2F));
      float adn = artanh_(dn);
      float m_[8];
#pragma unroll
      for (int j = 0; j < 8; ++j) m_[j] = zv[j] * delta[j];
      float md = rdot8(m_, m_);
      float mn = sqrtf(fmaxf(md, EPS2F));
      float sc = tanhf(mn/dn*adn)/mn;                // mobius_pointwise_mul(z, delta)
      float zd[8];
#pragma unroll
      for (int j = 0; j < 8; ++j) zd[j] = sc*m_[j];
      float zd2 = sc*sc*md;
      float hnew[8];
      mobius_add8(h, zd, h2, zd2, hnew);             // h (+) z*delta
#pragma unroll
      for (int j = 0; j < 8; ++j){
        h[j] = hnew[j];
        hbuf[row*HID + ((lane + 32*j) ^ sw)] = h[j];
        outp[rix*HID + lane + 32*j] = h[j];
      }
    }
    __syncthreads();
  }
#pragma unroll
  for (int j = 0; j < 8; ++j)
    hlast[(size_t)bg*HID + lane + 32*j] = h[j];
}

// ---------------- launcher ----------------
extern "C" void kernel_launch(void* const* d_in, const int* in_sizes, int n_in,
                              void* d_out, int out_size, void* d_ws, size_t ws_size,
                              hipStream_t stream){
  (void)in_sizes; (void)n_in; (void)out_size; (void)d_ws; (void)ws_size;
  const float* inp  = (const float*)d_in[0];   // [SEQ,BATCH,DIM]
  const float* h0   = (const float*)d_in[1];   // [NL,BATCH,HID]
  const float* wih  = (const float*)d_in[2];   // [NL,3*HID,DIM]
  const float* whh  = (const float*)d_in[3];   // [NL,3*HID,HID]
  const float* bias = (const float*)d_in[4];   // [NL,3,HID]
  float* outp = (float*)d_out;                          // [SEQ,BATCH,HID]
  float* ht   = outp + (size_t)SEQL*BATCH*HID;          // [NL,BATCH,HID]

  const int gemm_grid = (NROWS/16)*48/16;               // 12288 blocks
  // ---- layer 0 (expmap0 folded into projection epilogue)
  row_expmap_norms<<<NROWS/8, 256, 0, stream>>>(inp);
  ux_gemm<<<gemm_grid, 512, 0, stream>>>(inp, wih, 1);
  ux_norms<<<NROWS/8, 256, 0, stream>>>(nullptr);
  pack_whh<<<3*128*HID/256, 256, 0, stream>>>(whh);
  mobius_gru_scan<<<BATCH/16, 512, 0, stream>>>(h0, bias, outp, ht);
  // ---- layer 1 (consumes layer-0 outputs staged in d_out, then overwrites them)
  ux_gemm<<<gemm_grid, 512, 0, stream>>>(outp, wih + (size_t)G3*DIMS, 0);
  ux_norms<<<NROWS/8, 256, 0, stream>>>(outp);
  pack_whh<<<3*128*HID/256, 256, 0, stream>>>(whh + (size_t)G3*HID);
  mobius_gru_scan<<<BATCH/16, 512, 0, stream>>>(h0 + (size_t)BATCH*HID,
                                                bias + 3*HID,
                                                outp, ht + (size_t)BATCH*HID);
}